// TopKActivation_27152783245521
// MI455X (gfx1250) — compile-verified
//
#include <hip/hip_runtime.h>

#define COLS 32768
#define TPB  1024
#define TOPK 32u
#define NHIST 8  // histogram replicas (wave-interleaved) to tame ds_add conflicts
// LDS: row data (128KB) + 8x256 hist + 64 tie cols + eq_n + 2 broadcast words
#define SMEM_BYTES (COLS * 4 + (NHIST * 256) * 4 + 64 * 4 + 4 + 8)

// Vector types matching the async-to-LDS builtin / nontemporal store.
typedef int   v4i __attribute__((vector_size(16)));
typedef float v4f __attribute__((ext_vector_type(4)));
typedef __attribute__((address_space(1))) v4i* gv4i_ptr;  // global
typedef __attribute__((address_space(3))) v4i* lv4i_ptr;  // LDS

// Order-preserving map float -> uint (descending float == descending key)
__device__ __forceinline__ unsigned f2key(float f) {
  unsigned u = __float_as_uint(f);
  return (u & 0x80000000u) ? ~u : (u | 0x80000000u);
}

__global__ __launch_bounds__(TPB) void topk_mask_kernel(const float* __restrict__ x,
                                                        float* __restrict__ out) {
  extern __shared__ unsigned char smem[];
  float*    sdata   = reinterpret_cast<float*>(smem);
  unsigned* hist    = reinterpret_cast<unsigned*>(smem + COLS * 4);  // NHIST*256 u32
  unsigned* eq_cols = hist + NHIST * 256;                            // 64 u32
  unsigned* eq_n    = eq_cols + 64;                                  // 1 u32
  unsigned* bc      = eq_n + 1;                                      // 2 u32

  const unsigned t   = threadIdx.x;
  const unsigned row = blockIdx.x;
  const float* __restrict__ rowp = x   + (size_t)row * COLS;
  float*       __restrict__ outp = out + (size_t)row * COLS;

  // ---- Stage the whole 128KB row into LDS via CDNA5 async global->LDS ----
  // thread t owns columns {i*4096 + 4t .. +3}: each wave issues contiguous 512B,
  // each b128 instruction per block covers 16KB of contiguous HBM.
  // cpol=1 -> th:TH_NT: streamed once, keep it out of L2.
#pragma unroll
  for (int i = 0; i < 8; ++i) {
    const unsigned col = (unsigned)i * 4096u + t * 4u;
#if __has_builtin(__builtin_amdgcn_global_load_async_to_lds_b128)
    __builtin_amdgcn_global_load_async_to_lds_b128(
        (gv4i_ptr)(rowp + col),
        (lv4i_ptr)(sdata + col),
        /*offset=*/0, /*cpol=*/1);
#else
    *reinterpret_cast<float4*>(sdata + col) =
        *reinterpret_cast<const float4*>(rowp + col);
#endif
  }
#if __has_builtin(__builtin_amdgcn_s_wait_asynccnt)
  __builtin_amdgcn_s_wait_asynccnt(0);
#else
  asm volatile("s_wait_asynccnt 0" ::: "memory");
#endif
  __syncthreads();

  // ---- 4-round 8-bit radix select of the 32nd-largest key, all from LDS ----
  unsigned prefix = 0, dmask = 0, krem = TOPK;
  const unsigned hcopy = ((t >> 5) & (NHIST - 1u)) * 256u;  // per-wave replica

#pragma unroll 1
  for (int round = 0; round < 4; ++round) {
    const int shift = 24 - round * 8;
    hist[t] = 0;             // clears replicas 0..3
    hist[t + 1024] = 0;      // clears replicas 4..7
    __syncthreads();

#pragma unroll
    for (int i = 0; i < 8; ++i) {
      const unsigned col = (unsigned)i * 4096u + t * 4u;
      const float4 v = *reinterpret_cast<const float4*>(sdata + col);
      const float vv[4] = {v.x, v.y, v.z, v.w};
#pragma unroll
      for (int j = 0; j < 4; ++j) {
        const unsigned kk = f2key(vv[j]);
        if ((kk & dmask) == prefix)
          atomicAdd(&hist[hcopy + ((kk >> shift) & 0xFFu)], 1u);
      }
    }
    __syncthreads();

    if (t < 256u) {
      unsigned s = 0;
#pragma unroll
      for (int c = 0; c < NHIST; ++c) s += hist[t + c * 256];
      hist[t] = s;
    }
    __syncthreads();

    // Parallel suffix-sum over hist[0..255] (Hillis-Steele, double-buffered).
    unsigned* src = hist;
    unsigned* dst = hist + 256;
#pragma unroll 1
    for (unsigned ofs = 1; ofs < 256u; ofs <<= 1) {
      if (t < 256u) dst[t] = src[t] + ((t + ofs < 256u) ? src[t + ofs] : 0u);
      __syncthreads();
      unsigned* tmp = src; src = dst; dst = tmp;
    }
    if (t < 256u) {
      const unsigned s     = src[t];
      const unsigned snext = (t < 255u) ? src[t + 1] : 0u;
      if (s >= krem && snext < krem) { bc[0] = t; bc[1] = krem - snext; }
    }
    __syncthreads();
    prefix |= bc[0] << shift;
    dmask  |= 0xFFu << shift;
    krem    = bc[1];
    __syncthreads();
  }
  const unsigned T = prefix;  // exact key of the k-th largest element

  // ---- Ties: collect columns whose key == T (virtually always exactly 1) ----
  if (t == 0) *eq_n = 0;
  __syncthreads();
#pragma unroll
  for (int i = 0; i < 8; ++i) {
    const unsigned col = (unsigned)i * 4096u + t * 4u;
    const float4 v = *reinterpret_cast<const float4*>(sdata + col);
    const float vv[4] = {v.x, v.y, v.z, v.w};
#pragma unroll
    for (int j = 0; j < 4; ++j) {
      if (f2key(vv[j]) == T) {
        const unsigned idx = atomicAdd(eq_n, 1u);
        if (idx < 64u) eq_cols[idx] = col + (unsigned)j;
      }
    }
  }
  __syncthreads();
  const unsigned ne = (*eq_n < 64u) ? *eq_n : 64u;

  // ---- Masked output: coalesced non-temporal b128 stores (write-once data) ----
#pragma unroll
  for (int i = 0; i < 8; ++i) {
    const unsigned col = (unsigned)i * 4096u + t * 4u;
    const float4 v = *reinterpret_cast<const float4*>(sdata + col);
    const float vv[4] = {v.x, v.y, v.z, v.w};
    v4f o;
#pragma unroll
    for (int j = 0; j < 4; ++j) {
      const unsigned kk = f2key(vv[j]);
      bool pass = kk > T;
      if (kk == T) {  // admit equal values in column order (matches top_k ties)
        unsigned rank = 0;
        for (unsigned e = 0; e < ne; ++e)
          rank += (eq_cols[e] < col + (unsigned)j) ? 1u : 0u;
        pass = rank < krem;
      }
      o[j] = pass ? vv[j] : 0.0f;
    }
    __builtin_nontemporal_store(o, reinterpret_cast<v4f*>(outp + col));
  }
}

extern "C" void kernel_launch(void* const* d_in, const int* in_sizes, int n_in,
                              void* d_out, int out_size, void* d_ws, size_t ws_size,
                              hipStream_t stream) {
  (void)n_in; (void)d_ws; (void)ws_size; (void)out_size;
  const float* x = (const float*)d_in[0];
  float* out = (float*)d_out;
  const int rows = in_sizes[0] / COLS;

  // Allow >64KB dynamic LDS (CDNA5 WGP has 320KB; we use ~140KB per block).
  (void)hipFuncSetAttribute(reinterpret_cast<const void*>(topk_mask_kernel),
                            hipFuncAttributeMaxDynamicSharedMemorySize, SMEM_BYTES);

  topk_mask_kernel<<<rows, TPB, SMEM_BYTES, stream>>>(x, out);
}